// Embedding_32719060861121
// MI455X (gfx1250) — compile-verified
//
#include <hip/hip_runtime.h>
#include <hip/hip_bf16.h>

// Problem constants from the reference
#define S_LEN   512
#define B_COLS  512
#define D_DIM   256
#define SEP_ID  13
#define PAD_ID  1

typedef float v4f __attribute__((ext_vector_type(4)));

// ---------------------------------------------------------------------------
// Phase 1: per-column pad-aware position scan (inherently sequential in j).
// One lane per batch column b; lanes read consecutive int32s of each sequence
// row -> fully coalesced. Packs everything phase 2 needs into ONE u32:
//   ctrl = idx[9:0] | tok[21:12] | skip[31]
// (idx = j - pad_num <= 511, tok <= 511). ~2 MiB traffic total; the serial
// chain is ALU-only, loads are prefetched/pipelined ahead.
// ---------------------------------------------------------------------------
__global__ __launch_bounds__(256)
void pad_scan_kernel(const int* __restrict__ seq,
                     unsigned int* __restrict__ ctrl) {
    const int b = blockIdx.x * blockDim.x + threadIdx.x;
    if (b >= B_COLS) return;

    bool prev_sep = false;
    int  pad_num  = 0;

#pragma unroll 4
    for (int j = 0; j < S_LEN; ++j) {
        if ((j & 7) == 0 && (j + 8) < S_LEN) {
            // gfx1250 speculative prefetch of the row 8 steps ahead
            __builtin_prefetch(&seq[(size_t)(j + 8) * B_COLS + b], 0, 0);
        }
        const int tok = seq[(size_t)j * B_COLS + b];

        const bool is_sep = (tok == SEP_ID);
        const bool skip   = prev_sep && (tok == PAD_ID) && !is_sep;

        const unsigned int idx = (unsigned int)(j - pad_num);   // 0..511
        ctrl[(size_t)j * B_COLS + b] =
            idx | ((unsigned int)tok << 12) | (skip ? 0x80000000u : 0u);

        prev_sep = is_sep || skip;
        pad_num += skip ? 1 : 0;
    }
}

// ---------------------------------------------------------------------------
// Phase 2: fully parallel streaming gather+add. One wave32 per (j,b) row of
// 256 floats (1 KiB); each lane owns 8 floats.
//
// row is wave-uniform: readfirstlane pins it to an SGPR so the ctrl fetch
// becomes a single s_load_b32 (KMcnt, scalar cache — coherent here because
// the dispatch boundary between phase 1 and phase 2 invalidates it), the
// tok/idx/skip decode becomes SALU, and the W/PE/out accesses become
// saddr-form global_load_b128/global_store_b128 (SGPR base + lane offset).
//
// token_W row 0 is all-zero by construction, so the reference's (tok != 0)
// mask is implicit in the gather; skip folds into a 0/1 multiplier ->
// branch-free, divergence-free body. Output stores are NONTEMPORAL so the
// 256 MiB write stream does not evict the ~1 MiB L2-resident tables.
// ---------------------------------------------------------------------------
__global__ __launch_bounds__(256)
void embed_add_kernel(const unsigned int* __restrict__ ctrl,
                      const float* __restrict__ W,
                      const float* __restrict__ PE,
                      float* __restrict__ out) {
    const size_t gid = (size_t)blockIdx.x * blockDim.x + threadIdx.x;
    const int    row = __builtin_amdgcn_readfirstlane((int)(gid >> 5));
    const int    c   = (threadIdx.x & 31) * 8;     // 8 floats per lane

    const unsigned int ct = ctrl[row];             // uniform -> s_load_b32

    const int   tok = (int)((ct >> 12) & 0x3FFu);
    const int   idx = (int)(ct & 0xFFFu);
    const float pm  = (ct & 0x80000000u) ? 0.0f : 1.0f;

    const float* w = W  + (size_t)tok * D_DIM + c;
    const float* p = PE + (size_t)idx * D_DIM + c;

    const v4f t0 = *(const v4f*)(w);
    const v4f t1 = *(const v4f*)(w + 4);
    const v4f p0 = *(const v4f*)(p);
    const v4f p1 = *(const v4f*)(p + 4);

    const v4f r0 = t0 + pm * p0;
    const v4f r1 = t1 + pm * p1;

    float* o = out + (size_t)row * D_DIM + c;
    __builtin_nontemporal_store(r0, (v4f*)o);          // TH=NT stores
    __builtin_nontemporal_store(r1, (v4f*)(o + 4));
}

// ---------------------------------------------------------------------------
// Launch. Inputs (setup_inputs order): [0] sequence int32 [S,B],
// [1] token_W f32 [512,256], [2] pe f32 [512,256]. Output f32 [S,B,D].
// Workspace: 1 MiB ctrl array (S*B uint32), fully rewritten every call
// (deterministic; no state carried across calls).
// ---------------------------------------------------------------------------
extern "C" void kernel_launch(void* const* d_in, const int* in_sizes, int n_in,
                              void* d_out, int out_size, void* d_ws, size_t ws_size,
                              hipStream_t stream) {
    (void)in_sizes; (void)n_in; (void)out_size; (void)ws_size;

    const int*   seq = (const int*)d_in[0];
    const float* W   = (const float*)d_in[1];
    const float* PE  = (const float*)d_in[2];
    float*       out = (float*)d_out;

    unsigned int* ctrl = (unsigned int*)d_ws;  // S_LEN*B_COLS*4 = 1 MiB

    // Phase 1: 512 columns, 256 threads/block (8 wave32 per block)
    pad_scan_kernel<<<(B_COLS + 255) / 256, 256, 0, stream>>>(seq, ctrl);

    // Phase 2: one wave32 per row of 256 floats -> S*B*32 threads
    const size_t total_threads = (size_t)S_LEN * B_COLS * 32;
    const int    block         = 256;
    const int    grid          = (int)(total_threads / block);  // 32768
    embed_add_kernel<<<grid, block, 0, stream>>>(ctrl, W, PE, out);
}